// GraphSAGE_65377992179876
// MI455X (gfx1250) — compile-verified
//
#include <hip/hip_runtime.h>

typedef __attribute__((ext_vector_type(2))) float v2f;
typedef __attribute__((ext_vector_type(8))) float v8f;

#define WMMA_F32X4(a, b, c) \
  __builtin_amdgcn_wmma_f32_16x16x4_f32(false, (a), false, (b), (short)0, (c), false, false)

#define NN  768
#define HH  128
#define EIN 32
#define NSPLIT 4                 // sender-range splits for the fused kernel
#define CHUNKS_PER_SPLIT (NN / 8 / NSPLIT)

// ---------------------------------------------------------------------------
// Transpose: WT[n*K + k] = W[k*N + n]   (W is K x N row-major)
// ---------------------------------------------------------------------------
__global__ __launch_bounds__(256) void transpose_k(const float* __restrict__ W,
                                                   float* __restrict__ WT,
                                                   int K, int N)
{
  const int idx = blockIdx.x * 256 + threadIdx.x;
  if (idx < K * N) {
    const int n = idx / K;
    const int k = idx - n * K;
    WT[idx] = W[(size_t)k * N + n];
  }
}

// ---------------------------------------------------------------------------
// Generic 16x16-tile GEMM with TRANSPOSED weights:
// C = [relu](A @ W + bias), WT is (Ncols x K) so B pairs are contiguous b64.
// One wave per output tile.
// ---------------------------------------------------------------------------
__global__ __launch_bounds__(32) void wgemm_k(
    const float* __restrict__ A, const float* __restrict__ WT,
    const float* __restrict__ bias, float* __restrict__ C,
    int K, int Ncols, int do_relu)
{
  const int m0   = blockIdx.x * 16;
  const int n0   = blockIdx.y * 16;
  const int lane = threadIdx.x & 31;
  const int l16  = lane & 15;
  const int lhi  = lane >> 4;

  v8f acc = {};
  const float* arow = A  + (size_t)(m0 + l16) * K;
  const float* brow = WT + (size_t)(n0 + l16) * K;
  for (int k0 = 0; k0 < K; k0 += 4) {
    v2f a, b;
    a[0] = arow[k0 + 2 * lhi];
    a[1] = arow[k0 + 2 * lhi + 1];
    b[0] = brow[k0 + 2 * lhi];
    b[1] = brow[k0 + 2 * lhi + 1];
    acc = WMMA_F32X4(a, b, acc);
  }
  const float bv = bias ? bias[n0 + l16] : 0.0f;
#pragma unroll
  for (int r = 0; r < 8; ++r) {
    float v = acc[r] + bv;
    if (do_relu) v = v > 0.0f ? v : 0.0f;
    C[(size_t)(m0 + r + 8 * lhi) * Ncols + n0 + l16] = v;
  }
}

// ---------------------------------------------------------------------------
// Update GEMM: C = relu(A1 @ W1 + (sum_s aggP[s]) / deg @ W2 + bias)
// W1T/W2T transposed (128 x 128). Combines the NSPLIT partial agg buffers and
// the degree normalization on the fly (linear, so exact).
// ---------------------------------------------------------------------------
__global__ __launch_bounds__(32) void wgemm_update_k(
    const float* __restrict__ A1, const float* __restrict__ W1T,
    const float* __restrict__ aggP, const float* __restrict__ W2T,
    const float* __restrict__ bias, const float* __restrict__ deg,
    float* __restrict__ C)
{
  const int m0   = blockIdx.x * 16;
  const int n0   = blockIdx.y * 16;
  const int lane = threadIdx.x & 31;
  const int l16  = lane & 15;
  const int lhi  = lane >> 4;
  const int row  = m0 + l16;

  v8f acc = {};
  const float* a1  = A1  + (size_t)row * HH;
  const float* b1  = W1T + (size_t)(n0 + l16) * HH;
  const float* b2  = W2T + (size_t)(n0 + l16) * HH;
  const float* p0  = aggP + 0 * (size_t)NN * HH + (size_t)row * HH;
  const float* p1  = aggP + 1 * (size_t)NN * HH + (size_t)row * HH;
  const float* p2  = aggP + 2 * (size_t)NN * HH + (size_t)row * HH;
  const float* p3  = aggP + 3 * (size_t)NN * HH + (size_t)row * HH;
  const float rdeg = 1.0f / deg[row];

  for (int k0 = 0; k0 < HH; k0 += 4) {
    v2f a, b;
    a[0] = a1[k0 + 2 * lhi];
    a[1] = a1[k0 + 2 * lhi + 1];
    b[0] = b1[k0 + 2 * lhi];
    b[1] = b1[k0 + 2 * lhi + 1];
    acc = WMMA_F32X4(a, b, acc);
  }
  for (int k0 = 0; k0 < HH; k0 += 4) {
    const int k = k0 + 2 * lhi;
    v2f a, b;
    a[0] = (p0[k]     + p1[k]     + p2[k]     + p3[k])     * rdeg;
    a[1] = (p0[k + 1] + p1[k + 1] + p2[k + 1] + p3[k + 1]) * rdeg;
    b[0] = b2[k];
    b[1] = b2[k + 1];
    acc = WMMA_F32X4(a, b, acc);
  }
  const float bv = bias[n0 + l16];
#pragma unroll
  for (int r = 0; r < 8; ++r) {
    float v = acc[r] + bv;
    v = v > 0.0f ? v : 0.0f;
    C[(size_t)(m0 + r + 8 * lhi) * HH + n0 + l16] = v;
  }
}

// ---------------------------------------------------------------------------
// Fused per-round edge-message + partial aggregation kernel.
// grid = (48 j-tiles, NSPLIT sender-splits); block = 256 threads = 8 waves.
// Wave w of split s owns senders i = (s*CHUNKS + c)*8 + w.
// Writes (un-normalized) partial sums:
//   aggP[s][j,k] = sum_{i in split s} adj[i,j] * relu(sp[i,k]+eproj[i,j,k]+ba[k])
// eproj recomputed on the fly: relu(edge[i,j,:] @ We + be) @ Wa2.
// WeT staged in LDS; Wa2T read from global (L2-resident), contiguous b64 pairs.
// ---------------------------------------------------------------------------
__global__ __launch_bounds__(256) void edge_agg_k(
    const float* __restrict__ edgef, const float* __restrict__ adj,
    const float* __restrict__ sp,    const float* __restrict__ WeT,
    const float* __restrict__ be,    const float* __restrict__ Wa2T,
    const float* __restrict__ ba,
    float* __restrict__ aggP)
{
  __shared__ float Es[8][16][EIN];     // per-wave edge tile        (16 KB)
  __shared__ float Eh[8][16][32];      // per-wave e quarter tile   (16 KB)
  __shared__ float WeTs[HH][EIN];      // transposed We, block-wide (16 KB)
  __shared__ float aggS[16][HH];       // block reduction            (8 KB)

  const int j0   = blockIdx.x * 16;
  const int spl  = blockIdx.y;
  const int tid  = threadIdx.x;
  const int w    = tid >> 5;
  const int lane = tid & 31;
  const int l16  = lane & 15;
  const int lhi  = lane >> 4;

  // Stage WeT (128x32) once per block, coalesced.
  {
    const float4* src = (const float4*)WeT;
    float4*       dst = (float4*)&WeTs[0][0];
#pragma unroll
    for (int q = 0; q < 4; ++q) dst[tid + 256 * q] = src[tid + 256 * q];
  }
  __syncthreads();

  v8f gacc[8] = {};   // per-wave agg accumulators (rows = j_local, cols = k)

  for (int c = 0; c < CHUNKS_PER_SPLIT; ++c) {
    const int i = (spl * CHUNKS_PER_SPLIT + c) * 8 + w;   // sender for this wave

    // Stage this sender's 16x32 edge tile (contiguous 512 floats) into LDS.
    {
      const float4* src = (const float4*)(edgef + ((size_t)i * NN + j0) * EIN);
      float4*       dst = (float4*)&Es[w][0][0];
#pragma unroll
      for (int q = 0; q < 4; ++q) dst[lane + 32 * q] = src[lane + 32 * q];
    }

    v8f pacc[8] = {};   // e_proj tile accumulators: rows j_local, cols 128

#pragma unroll
    for (int q = 0; q < 4; ++q) {        // quarter of the H dimension (32 cols)
      // GEMM1: Eh = relu(Es(16x32) @ We[:, 32q .. 32q+32] + be)
#pragma unroll
      for (int nt = 0; nt < 2; ++nt) {
        const int ncol = q * 32 + nt * 16 + l16;     // global e-channel
        v8f e = {};
#pragma unroll
        for (int k0 = 0; k0 < EIN; k0 += 4) {
          v2f a, b;
          a[0] = Es[w][l16][k0 + 2 * lhi];
          a[1] = Es[w][l16][k0 + 2 * lhi + 1];
          b[0] = WeTs[ncol][k0 + 2 * lhi];
          b[1] = WeTs[ncol][k0 + 2 * lhi + 1];
          e = WMMA_F32X4(a, b, e);
        }
        const float bev = be[ncol];
#pragma unroll
        for (int r = 0; r < 8; ++r) {
          float v = e[r] + bev;
          v = v > 0.0f ? v : 0.0f;
          Eh[w][r + 8 * lhi][nt * 16 + l16] = v;   // same-wave LDS round-trip
        }
      }
      // GEMM2 partial: pacc += Eh(16x32) @ Wa2[32q .. 32q+32, :]
#pragma unroll
      for (int nt2 = 0; nt2 < 8; ++nt2) {
        const float* brow = Wa2T + (size_t)(nt2 * 16 + l16) * HH + q * 32;
        v8f acc = pacc[nt2];
#pragma unroll
        for (int k0 = 0; k0 < 32; k0 += 4) {
          v2f a, b;
          a[0] = Eh[w][l16][k0 + 2 * lhi];
          a[1] = Eh[w][l16][k0 + 2 * lhi + 1];
          b[0] = brow[k0 + 2 * lhi];
          b[1] = brow[k0 + 2 * lhi + 1];
          acc = WMMA_F32X4(a, b, acc);
        }
        pacc[nt2] = acc;
      }
    }

    // Epilogue: relu(P + sp[i] + ba) * adj[i,j], accumulate into gacc.
    const float4 av0 = *(const float4*)(adj + (size_t)i * NN + j0 + 8 * lhi);
    const float4 av1 = *(const float4*)(adj + (size_t)i * NN + j0 + 8 * lhi + 4);
    const float adjv[8] = {av0.x, av0.y, av0.z, av0.w, av1.x, av1.y, av1.z, av1.w};
#pragma unroll
    for (int nt2 = 0; nt2 < 8; ++nt2) {
      const int col  = nt2 * 16 + l16;
      const float sv = sp[(size_t)i * HH + col] + ba[col];
#pragma unroll
      for (int r = 0; r < 8; ++r) {
        float v = pacc[nt2][r] + sv;
        v = v > 0.0f ? v : 0.0f;
        gacc[nt2][r] += v * adjv[r];
      }
    }
  }

  // Cross-wave reduction into aggS (deterministic 8-phase), then store partial.
  for (int idx = tid; idx < 16 * HH; idx += 256) ((float*)aggS)[idx] = 0.0f;
  __syncthreads();
  for (int p = 0; p < 8; ++p) {
    if (w == p) {
#pragma unroll
      for (int nt2 = 0; nt2 < 8; ++nt2)
#pragma unroll
        for (int r = 0; r < 8; ++r)
          aggS[r + 8 * lhi][nt2 * 16 + l16] += gacc[nt2][r];
    }
    __syncthreads();
  }
  float* out = aggP + (size_t)spl * NN * HH;
  for (int idx = tid; idx < 16 * HH; idx += 256) {
    const int jl = idx >> 7;
    const int k  = idx & (HH - 1);
    out[(size_t)(j0 + jl) * HH + k] = aggS[jl][k];
  }
}

// ---------------------------------------------------------------------------
// deg[j] = max(1, sum_i adj[i,j])
// ---------------------------------------------------------------------------
__global__ __launch_bounds__(256) void deg_k(const float* __restrict__ adj,
                                             float* __restrict__ deg)
{
  const int j = blockIdx.x * 256 + threadIdx.x;
  if (j < NN) {
    float s = 0.0f;
    for (int i = 0; i < NN; ++i) s += adj[(size_t)i * NN + j];
    deg[j] = s < 1.0f ? 1.0f : s;
  }
}

// ---------------------------------------------------------------------------
// graph_emb = mean over rows of h
// ---------------------------------------------------------------------------
__global__ __launch_bounds__(128) void emb_k(const float* __restrict__ h,
                                             float* __restrict__ emb)
{
  const int k = threadIdx.x;
  float s = 0.0f;
  for (int r = 0; r < NN; ++r) s += h[(size_t)r * HH + k];
  emb[k] = s * (1.0f / (float)NN);
}

// ---------------------------------------------------------------------------
extern "C" void kernel_launch(void* const* d_in, const int* in_sizes, int n_in,
                              void* d_out, int out_size, void* d_ws, size_t ws_size,
                              hipStream_t stream)
{
  (void)in_sizes; (void)n_in; (void)out_size; (void)ws_size;

  const float* node  = (const float*)d_in[0];
  const float* edgef = (const float*)d_in[1];
  const float* adj   = (const float*)d_in[2];
  const float* Wn    = (const float*)d_in[3];
  const float* bn    = (const float*)d_in[4];
  const float* We    = (const float*)d_in[5];
  const float* be    = (const float*)d_in[6];
  const float* Wa    = (const float*)d_in[7];
  const float* ba    = (const float*)d_in[8];
  const float* Wu    = (const float*)d_in[9];
  const float* bu    = (const float*)d_in[10];
  float* out = (float*)d_out;

  float* ws    = (float*)d_ws;
  float* hA    = ws;
  float* hB    = hA   + (size_t)NN * HH;
  float* sp    = hB   + (size_t)NN * HH;
  float* aggP  = sp   + (size_t)NN * HH;                 // NSPLIT partials
  float* WnT   = aggP + (size_t)NSPLIT * NN * HH;        // 128 x 64
  float* WaT1  = WnT  + (size_t)HH * 64;                 // 128 x 128
  float* Wa2T  = WaT1 + (size_t)HH * HH;                 // 128 x 128
  float* WuT1  = Wa2T + (size_t)HH * HH;                 // 128 x 128
  float* WuT2  = WuT1 + (size_t)HH * HH;                 // 128 x 128
  float* WeT   = WuT2 + (size_t)HH * HH;                 // 128 x 32
  float* degw  = WeT  + (size_t)HH * EIN;

  const dim3 gemmGrid(NN / 16, HH / 16);

  // Weight transposes (tiny, once per call; deterministic).
  transpose_k<<<dim3(32), dim3(256), 0, stream>>>(Wn,            WnT,  64, HH);
  transpose_k<<<dim3(64), dim3(256), 0, stream>>>(Wa,            WaT1, HH, HH);
  transpose_k<<<dim3(64), dim3(256), 0, stream>>>(Wa + HH * HH,  Wa2T, HH, HH);
  transpose_k<<<dim3(64), dim3(256), 0, stream>>>(Wu,            WuT1, HH, HH);
  transpose_k<<<dim3(64), dim3(256), 0, stream>>>(Wu + HH * HH,  WuT2, HH, HH);
  transpose_k<<<dim3(16), dim3(256), 0, stream>>>(We,            WeT, EIN, HH);

  deg_k<<<dim3(3), dim3(256), 0, stream>>>(adj, degw);

  // h0 = relu(node @ Wn + bn), K = 64
  wgemm_k<<<gemmGrid, dim3(32), 0, stream>>>(node, WnT, bn, hA, 64, HH, 1);

  // ---- round 0 ----
  wgemm_k<<<gemmGrid, dim3(32), 0, stream>>>(hA, WaT1, nullptr, sp, HH, HH, 0);
  edge_agg_k<<<dim3(NN / 16, NSPLIT), dim3(256), 0, stream>>>(
      edgef, adj, sp, WeT, be, Wa2T, ba, aggP);
  wgemm_update_k<<<gemmGrid, dim3(32), 0, stream>>>(hA, WuT1, aggP, WuT2, bu,
                                                    degw, hB);

  // ---- round 1 (writes h straight into d_out) ----
  wgemm_k<<<gemmGrid, dim3(32), 0, stream>>>(hB, WaT1, nullptr, sp, HH, HH, 0);
  edge_agg_k<<<dim3(NN / 16, NSPLIT), dim3(256), 0, stream>>>(
      edgef, adj, sp, WeT, be, Wa2T, ba, aggP);
  wgemm_update_k<<<gemmGrid, dim3(32), 0, stream>>>(hB, WuT1, aggP, WuT2, bu,
                                                    degw, out);

  // graph_emb = h.mean(axis=0)
  emb_k<<<dim3(1), dim3(128), 0, stream>>>(out, out + (size_t)NN * HH);
}